// LayerNormIndRNN_14181982012202
// MI455X (gfx1250) — compile-verified
//
#include <hip/hip_runtime.h>

typedef __attribute__((ext_vector_type(2))) float v2f;
typedef __attribute__((ext_vector_type(8))) float v8f;

#define S_LEN 1024
#define BATCH 64
#define HID   512
#define KDIM  512
#define MROWS (S_LEN * BATCH)
#define LDS_STRIDE 68   // 64 + 4 floats pad: bank-conflict-free, keeps 16B alignment
#define NSLAB (KDIM / 64)

// One 16-byte async global->LDS copy (per lane). ASYNCcnt-tracked, no VGPR data.
__device__ __forceinline__ void async_copy_b128(const float* gptr, const float* lptr) {
  const unsigned int       lds  = (unsigned int)(unsigned long long)lptr; // addr[31:0] = LDS offset
  const unsigned long long gadr = (unsigned long long)gptr;
  asm volatile("global_load_async_to_lds_b128 %0, %1, off"
               :: "v"(lds), "v"(gadr) : "memory");
}
// ASYNCcnt decrements in order: <=12 outstanding => previous slab's 12 copies done.
__device__ __forceinline__ void wait_async_prev() {
  asm volatile("s_wait_asynccnt 12" ::: "memory");
}
__device__ __forceinline__ void wait_async_all() {
  asm volatile("s_wait_asynccnt 0" ::: "memory");
}

// Issue all 12 async copies (8 A-rows + 4 W-rows worth of float4s) for one K-slab.
__device__ __forceinline__ void stage_slab(const float* __restrict__ A,
                                           const float* __restrict__ W,
                                           int mBase, int nBase, int kb,
                                           float* as, float* bs, int tid) {
#pragma unroll
  for (int r = 0; r < 8; ++r) {           // A: 128 rows x 16 float4
    const int idx = tid + r * 256;
    const int row = idx >> 4;             // 0..127
    const int c4  = (idx & 15) << 2;      // 0,4,...,60
    async_copy_b128(A + (size_t)(mBase + row) * KDIM + kb + c4,
                    as + row * LDS_STRIDE + c4);
  }
#pragma unroll
  for (int r = 0; r < 4; ++r) {           // W: 64 rows x 16 float4
    const int idx = tid + r * 256;
    const int row = idx >> 4;             // 0..63
    const int c4  = (idx & 15) << 2;
    async_copy_b128(W + (size_t)(nBase + row) * KDIM + kb + c4,
                    bs + row * LDS_STRIDE + c4);
  }
}

// ---------------------------------------------------------------------------
// GEMM: C[M x 512] = A[M x 512] * W[512 x 512]^T   (W row-major (N,K))
// WG = 256 threads (8 waves), 128x64 output tile, K tiled by 64.
// Double-buffered LDS: slab i+1 streams in via async-to-LDS while slab i is
// consumed by v_wmma_f32_16x16x4_f32 (wave: 16x64 strip, 4 accums, A reuse x4).
// ---------------------------------------------------------------------------
__global__ __launch_bounds__(256)
void gemm_f32_wmma(const float* __restrict__ A, const float* __restrict__ W,
                   float* __restrict__ C) {
  __shared__ float As[2][128 * LDS_STRIDE];   // 2 x 34,816 B
  __shared__ float Bs[2][64 * LDS_STRIDE];    // 2 x 17,408 B  (total ~104 KB)

  const int tid   = threadIdx.x;
  const int lane  = tid & 31;
  const int wave  = tid >> 5;
  const int mBase = blockIdx.x * 128;
  const int nBase = blockIdx.y * 64;
  const int lm    = lane & 15;
  const int kk    = (lane < 16) ? 0 : 2;  // fp32 frag: lanes 16-31 hold K+2,K+3

  const v8f vzero = {0.f, 0.f, 0.f, 0.f, 0.f, 0.f, 0.f, 0.f};
  v8f acc[4] = {vzero, vzero, vzero, vzero};

  // Prologue: stage slab 0.
  stage_slab(A, W, mBase, nBase, 0, As[0], Bs[0], tid);

  for (int i = 0; i < NSLAB; ++i) {
    const int cur = i & 1;
    if (i + 1 < NSLAB) {
      // Buffer cur^1 was last read in iteration i-1; end-of-(i-1) barrier protects it.
      stage_slab(A, W, mBase, nBase, (i + 1) * 64, As[cur ^ 1], Bs[cur ^ 1], tid);
      wait_async_prev();   // slab i's copies have landed; slab i+1 keeps streaming
    } else {
      wait_async_all();
    }
    // Warm L2 two slabs ahead (one ahead of the async engine).
    if (i + 2 < NSLAB) {
      const int prow = tid >> 1;
      const int pcol = (tid & 1) << 5;
      __builtin_prefetch(A + (size_t)(mBase + prow) * KDIM + (i + 2) * 64 + pcol, 0, 1);
    }
    __syncthreads();       // every wave's slab-i copies visible to all waves

    const float* Ar = As[cur] + (wave * 16 + lm) * LDS_STRIDE + kk;
    const float* Bc = Bs[cur] + lm * LDS_STRIDE + kk;
#pragma unroll
    for (int k0 = 0; k0 < 64; k0 += 4) {
      const v2f a = *(const v2f*)(Ar + k0);
#pragma unroll
      for (int t = 0; t < 4; ++t) {
        const v2f bt = *(const v2f*)(Bc + t * 16 * LDS_STRIDE + k0);
        acc[t] = __builtin_amdgcn_wmma_f32_16x16x4_f32(false, a, false, bt,
                                                       (short)0, acc[t], false, false);
      }
    }
    __syncthreads();       // all waves done reading buf[cur] before it is restaged
  }

  // C/D layout: VGPR r -> row +r (lanes 0-15) / +8+r (lanes 16-31), col = lane%16
  const int mOff = mBase + wave * 16 + ((lane < 16) ? 0 : 8);
  const int nOff = nBase + lm;
#pragma unroll
  for (int t = 0; t < 4; ++t) {
#pragma unroll
    for (int r = 0; r < 8; ++r) {
      C[(size_t)(mOff + r) * HID + nOff + t * 16] = acc[t][r];
    }
  }
}

// ---------------------------------------------------------------------------
// IndRNN scan: one block per batch row b; 256 threads hold 2 hidden units each.
// h = relu6( LN( lin_t + bias + rec*h ) ), carried in registers over S steps.
// ---------------------------------------------------------------------------
__global__ __launch_bounds__(256)
void indrnn_scan(const float* __restrict__ lin, const float* __restrict__ bias,
                 const float* __restrict__ rec, const float* __restrict__ gamma,
                 const float* __restrict__ beta, float* __restrict__ y,
                 float* __restrict__ hid) {
  __shared__ float redS[8];
  __shared__ float redQ[8];
  __shared__ float statMean;
  __shared__ float statRstd;

  const int b    = blockIdx.x;
  const int tid  = threadIdx.x;
  const int lane = tid & 31;
  const int wave = tid >> 5;
  const int h0   = tid * 2;

  const float bx = bias[h0],  by = bias[h0 + 1];
  const float rx = rec[h0],   ry = rec[h0 + 1];
  const float gx = gamma[h0], gy = gamma[h0 + 1];
  const float ex = beta[h0],  ey = beta[h0 + 1];

  float hx = 0.f, hy = 0.f;

  for (int s = 0; s < S_LEN; ++s) {
    const size_t base = ((size_t)s * BATCH + b) * HID + h0;
    const float2 l2 = *(const float2*)(lin + base);
    float vx = l2.x + bx + rx * hx;
    float vy = l2.y + by + ry * hy;

    float s1 = vx + vy;
    float s2 = vx * vx + vy * vy;
#pragma unroll
    for (int off = 16; off > 0; off >>= 1) {
      s1 += __shfl_xor(s1, off, 32);
      s2 += __shfl_xor(s2, off, 32);
    }
    if (lane == 0) { redS[wave] = s1; redQ[wave] = s2; }
    __syncthreads();
    if (tid == 0) {
      float t1 = 0.f, t2 = 0.f;
#pragma unroll
      for (int i = 0; i < 8; ++i) { t1 += redS[i]; t2 += redQ[i]; }
      const float mean = t1 * (1.0f / HID);
      const float var  = t2 * (1.0f / HID) - mean * mean;
      statMean = mean;
      statRstd = __frsqrt_rn(var + 1e-6f);
    }
    __syncthreads();
    const float mean = statMean, rstd = statRstd;

    vx = (vx - mean) * rstd * gx + ex;
    vy = (vy - mean) * rstd * gy + ey;
    vx = fminf(fmaxf(vx, 0.f), 6.f);
    vy = fminf(fmaxf(vy, 0.f), 6.f);
    hx = vx; hy = vy;

    float2 o; o.x = vx; o.y = vy;
    *(float2*)(y + base) = o;
  }

  // hiddens layout after cat(-1)+reshape: flat[b*1024 + layer*512 + h]
  hid[(size_t)b * (2 * HID) + h0]     = hx;
  hid[(size_t)b * (2 * HID) + h0 + 1] = hy;
}

// ---------------------------------------------------------------------------
extern "C" void kernel_launch(void* const* d_in, const int* in_sizes, int n_in,
                              void* d_out, int out_size, void* d_ws, size_t ws_size,
                              hipStream_t stream) {
  const float* x    = (const float*)d_in[0];
  const float* W0   = (const float*)d_in[1];
  const float* b0   = (const float*)d_in[2];
  const float* rec0 = (const float*)d_in[3];
  const float* g0   = (const float*)d_in[4];
  const float* be0  = (const float*)d_in[5];
  const float* W1   = (const float*)d_in[6];
  const float* b1   = (const float*)d_in[7];
  const float* rec1 = (const float*)d_in[8];
  const float* g1   = (const float*)d_in[9];
  const float* be1  = (const float*)d_in[10];

  float* out     = (float*)d_out;
  float* yOut    = out;                                   // (S,B,H)
  float* hidTail = out + (size_t)S_LEN * BATCH * HID;     // (B, 2H) flat

  float* bufA = (float*)d_ws;                             // lin0 -> y0 (in place)
  float* bufB = bufA + (size_t)S_LEN * BATCH * HID;       // lin1

  dim3 gg(MROWS / 128, HID / 64), gb(256);
  dim3 sg(BATCH), sb(256);

  // Layer 0
  gemm_f32_wmma<<<gg, gb, 0, stream>>>(x, W0, bufA);
  indrnn_scan<<<sg, sb, 0, stream>>>(bufA, b0, rec0, g0, be0, bufA, hidTail + 0);
  // Layer 1
  gemm_f32_wmma<<<gg, gb, 0, stream>>>(bufA, W1, bufB);
  indrnn_scan<<<sg, sb, 0, stream>>>(bufB, b1, rec1, g1, be1, yOut, hidTail + HID);
}